// Top2SharedExpertMoE_77386720740075
// MI455X (gfx1250) — compile-verified
//
#include <hip/hip_runtime.h>
#include <hip/hip_bf16.h>
#include <math.h>
#include <float.h>

#define NEXP 8
#define HDIM 768
#define IDIM 3072
#define NTOK 4096
#define CAP  1280

typedef float v2f __attribute__((ext_vector_type(2)));
typedef float v8f __attribute__((ext_vector_type(8)));

// ---------------------------------------------------------------------------
// Router: logits -> softmax -> top2 -> normalized weights -> per-expert score
// score[e][n] = weight if expert e is in token n's top-2 else -inf
// ---------------------------------------------------------------------------
__global__ __launch_bounds__(256)
void router_kernel(const float* __restrict__ x,
                   const float* __restrict__ Wr,
                   float* __restrict__ score) {
    int n = blockIdx.x * blockDim.x + threadIdx.x;
    if (n >= NTOK) return;
    float acc[NEXP];
#pragma unroll
    for (int e = 0; e < NEXP; ++e) acc[e] = 0.f;
    const float* xr = x + (size_t)n * HDIM;
    for (int h = 0; h < HDIM; ++h) {
        float xv = xr[h];
#pragma unroll
        for (int e = 0; e < NEXP; ++e) acc[e] += xv * Wr[h * NEXP + e];
    }
    float mx = acc[0];
#pragma unroll
    for (int e = 1; e < NEXP; ++e) mx = fmaxf(mx, acc[e]);
    float p[NEXP]; float sum = 0.f;
#pragma unroll
    for (int e = 0; e < NEXP; ++e) { p[e] = __expf(acc[e] - mx); sum += p[e]; }
    float inv = 1.f / sum;
#pragma unroll
    for (int e = 0; e < NEXP; ++e) p[e] *= inv;
    // top-2, first occurrence wins ties (matches lax.top_k)
    int i1 = 0; float p1 = p[0];
#pragma unroll
    for (int e = 1; e < NEXP; ++e) if (p[e] > p1) { p1 = p[e]; i1 = e; }
    int i2 = -1; float p2 = -1.f;
#pragma unroll
    for (int e = 0; e < NEXP; ++e) if (e != i1 && p[e] > p2) { p2 = p[e]; i2 = e; }
    float denom = fmaxf(p1 + p2, 1e-9f);
    float w1v = p1 / denom, w2v = p2 / denom;
#pragma unroll
    for (int e = 0; e < NEXP; ++e) {
        float sc = -INFINITY;
        if (e == i1) sc = w1v; else if (e == i2) sc = w2v;
        score[(size_t)e * NTOK + n] = sc;
    }
}

// ---------------------------------------------------------------------------
// Capacity selection: exact stable argsort(-score)[:CAP] via rank counting.
// ---------------------------------------------------------------------------
__global__ __launch_bounds__(256)
void select_kernel(const float* __restrict__ score,
                   int* __restrict__ order,
                   float* __restrict__ wsel) {
    __shared__ float sc[NTOK];
    int e = blockIdx.y;
    const float* row = score + (size_t)e * NTOK;
    for (int i = threadIdx.x; i < NTOK; i += blockDim.x) sc[i] = row[i];
    __syncthreads();
    int n = blockIdx.x * blockDim.x + threadIdx.x;
    float s = sc[n];
    if (s > -FLT_MAX) {
        int rank = 0;
        for (int j = 0; j < NTOK; ++j) {
            float sj = sc[j];
            rank += (sj > s) || (sj == s && j < n);
        }
        if (rank < CAP) {
            order[e * CAP + rank] = n;
            wsel[e * CAP + rank] = s;
        }
    }
}

// ---------------------------------------------------------------------------
// WMMA GEMMs: block tile 64x128, BK=32, 8 waves, wave tile 32x32 (2x2 WMMA),
// software pipelined: global loads for chunk k+1 issue while chunk k computes.
// ---------------------------------------------------------------------------
#define SA 34
#define SB 34

__global__ __launch_bounds__(256)
void gateup_kernel(const float* __restrict__ A,
                   const int* __restrict__ rowidx,   // null => identity rows
                   const float* __restrict__ W1,     // [HDIM][IDIM]
                   const float* __restrict__ W3,     // [HDIM][IDIM]
                   float* __restrict__ hbuf) {       // [rows][IDIM]
    __shared__ float As[64 * SA];
    __shared__ float B1s[128 * SB];
    __shared__ float B3s[128 * SB];

    const int tid  = threadIdx.x;
    const int m0   = blockIdx.y * 64;
    const int n0   = blockIdx.x * 128;
    const int lane = tid & 31;
    const int wid  = tid >> 5;
    const int wm   = wid & 1;
    const int wn   = wid >> 1;
    const int half = lane >> 4;
    const int l15  = lane & 15;

    // staging geometry (loop invariant)
    const int ar = tid >> 2;                 // A row 0..63
    const int ac = (tid & 3) * 8;            // A col 0,8,16,24
    const int grow = rowidx ? rowidx[m0 + ar] : (m0 + ar);
    const float* aRow = A + (size_t)grow * HDIM + ac;
    int bkr[4], bnc[4];
#pragma unroll
    for (int i = 0; i < 4; ++i) {
        int lin = tid + i * 256;             // 0..1023 float4s
        bkr[i] = lin >> 5;                   // 0..31
        bnc[i] = (lin & 31) * 4;             // 0..124
    }

    v8f accg[2][2], accu[2][2];
#pragma unroll
    for (int i = 0; i < 2; ++i)
#pragma unroll
        for (int j = 0; j < 2; ++j) { accg[i][j] = (v8f)0.f; accu[i][j] = (v8f)0.f; }

    // prologue: load chunk 0 into registers
    float4 ra0, ra1, rb1[4], rb3[4];
    ra0 = *(const float4*)(aRow + 0);
    ra1 = *(const float4*)(aRow + 4);
#pragma unroll
    for (int i = 0; i < 4; ++i) {
        size_t goff = (size_t)bkr[i] * IDIM + n0 + bnc[i];
        rb1[i] = *(const float4*)(W1 + goff);
        rb3[i] = *(const float4*)(W3 + goff);
    }

    for (int k0 = 0; k0 < HDIM; k0 += 32) {
        // commit current chunk's registers to LDS
        {
            float* dst = &As[ar * SA + ac];
            dst[0] = ra0.x; dst[1] = ra0.y; dst[2] = ra0.z; dst[3] = ra0.w;
            dst[4] = ra1.x; dst[5] = ra1.y; dst[6] = ra1.z; dst[7] = ra1.w;
        }
#pragma unroll
        for (int i = 0; i < 4; ++i) {
            int kr = bkr[i], nc4 = bnc[i];
            B1s[(nc4 + 0) * SB + kr] = rb1[i].x; B1s[(nc4 + 1) * SB + kr] = rb1[i].y;
            B1s[(nc4 + 2) * SB + kr] = rb1[i].z; B1s[(nc4 + 3) * SB + kr] = rb1[i].w;
            B3s[(nc4 + 0) * SB + kr] = rb3[i].x; B3s[(nc4 + 1) * SB + kr] = rb3[i].y;
            B3s[(nc4 + 2) * SB + kr] = rb3[i].z; B3s[(nc4 + 3) * SB + kr] = rb3[i].w;
        }
        __syncthreads();

        // issue next chunk's global loads (in flight during WMMA below)
        int kn = k0 + 32;
        if (kn < HDIM) {
            ra0 = *(const float4*)(aRow + kn);
            ra1 = *(const float4*)(aRow + kn + 4);
#pragma unroll
            for (int i = 0; i < 4; ++i) {
                size_t goff = (size_t)(kn + bkr[i]) * IDIM + n0 + bnc[i];
                rb1[i] = *(const float4*)(W1 + goff);
                rb3[i] = *(const float4*)(W3 + goff);
            }
        }

        // compute current chunk from LDS
#pragma unroll
        for (int kk = 0; kk < 32; kk += 4) {
            v2f a[2], b1f[2], b3f[2];
#pragma unroll
            for (int tm = 0; tm < 2; ++tm) {
                int m = wm * 32 + tm * 16 + l15;
                a[tm] = *(const v2f*)&As[m * SA + kk + 2 * half];
            }
#pragma unroll
            for (int tn = 0; tn < 2; ++tn) {
                int nn = wn * 32 + tn * 16 + l15;
                b1f[tn] = *(const v2f*)&B1s[nn * SB + kk + 2 * half];
                b3f[tn] = *(const v2f*)&B3s[nn * SB + kk + 2 * half];
            }
#pragma unroll
            for (int tm = 0; tm < 2; ++tm)
#pragma unroll
                for (int tn = 0; tn < 2; ++tn) {
                    accg[tm][tn] = __builtin_amdgcn_wmma_f32_16x16x4_f32(
                        false, a[tm], false, b1f[tn], (short)0, accg[tm][tn], false, false);
                    accu[tm][tn] = __builtin_amdgcn_wmma_f32_16x16x4_f32(
                        false, a[tm], false, b3f[tn], (short)0, accu[tm][tn], false, false);
                }
        }
        __syncthreads();
    }
    // epilogue: SwiGLU, store h
#pragma unroll
    for (int tm = 0; tm < 2; ++tm)
#pragma unroll
        for (int tn = 0; tn < 2; ++tn) {
            int col = n0 + wn * 32 + tn * 16 + l15;
#pragma unroll
            for (int r = 0; r < 8; ++r) {
                int row = m0 + wm * 32 + tm * 16 + half * 8 + r;
                float g = accg[tm][tn][r];
                float u = accu[tm][tn][r];
                float h = (g / (1.f + __expf(-g))) * u;
                hbuf[(size_t)row * IDIM + col] = h;
            }
        }
}

// ---------------------------------------------------------------------------
// Down GEMM: out[row] (+)= (h @ W2)[m] * rowscale[m], scatter via rowidx.
// ---------------------------------------------------------------------------
__global__ __launch_bounds__(256)
void down_kernel(const float* __restrict__ hbuf,     // [rows][IDIM]
                 const float* __restrict__ W2,       // [IDIM][HDIM]
                 const int* __restrict__ rowidx,     // null => identity
                 const float* __restrict__ rowscale, // null => 1.0
                 float* __restrict__ out,            // [.][HDIM]
                 int use_atomic) {
    __shared__ float As[64 * SA];
    __shared__ float Bs[128 * SB];

    const int tid  = threadIdx.x;
    const int m0   = blockIdx.y * 64;
    const int n0   = blockIdx.x * 128;
    const int lane = tid & 31;
    const int wid  = tid >> 5;
    const int wm   = wid & 1;
    const int wn   = wid >> 1;
    const int half = lane >> 4;
    const int l15  = lane & 15;

    const int ar = tid >> 2;
    const int ac = (tid & 3) * 8;
    const float* aRow = hbuf + (size_t)(m0 + ar) * IDIM + ac;
    int bkr[4], bnc[4];
#pragma unroll
    for (int i = 0; i < 4; ++i) {
        int lin = tid + i * 256;
        bkr[i] = lin >> 5;
        bnc[i] = (lin & 31) * 4;
    }

    v8f acc[2][2];
#pragma unroll
    for (int i = 0; i < 2; ++i)
#pragma unroll
        for (int j = 0; j < 2; ++j) acc[i][j] = (v8f)0.f;

    float4 ra0, ra1, rb[4];
    ra0 = *(const float4*)(aRow + 0);
    ra1 = *(const float4*)(aRow + 4);
#pragma unroll
    for (int i = 0; i < 4; ++i)
        rb[i] = *(const float4*)(W2 + (size_t)bkr[i] * HDIM + n0 + bnc[i]);

    for (int k0 = 0; k0 < IDIM; k0 += 32) {
        {
            float* dst = &As[ar * SA + ac];
            dst[0] = ra0.x; dst[1] = ra0.y; dst[2] = ra0.z; dst[3] = ra0.w;
            dst[4] = ra1.x; dst[5] = ra1.y; dst[6] = ra1.z; dst[7] = ra1.w;
        }
#pragma unroll
        for (int i = 0; i < 4; ++i) {
            int kr = bkr[i], nc4 = bnc[i];
            Bs[(nc4 + 0) * SB + kr] = rb[i].x; Bs[(nc4 + 1) * SB + kr] = rb[i].y;
            Bs[(nc4 + 2) * SB + kr] = rb[i].z; Bs[(nc4 + 3) * SB + kr] = rb[i].w;
        }
        __syncthreads();

        int kn = k0 + 32;
        if (kn < IDIM) {
            ra0 = *(const float4*)(aRow + kn);
            ra1 = *(const float4*)(aRow + kn + 4);
#pragma unroll
            for (int i = 0; i < 4; ++i)
                rb[i] = *(const float4*)(W2 + (size_t)(kn + bkr[i]) * HDIM + n0 + bnc[i]);
        }

#pragma unroll
        for (int kk = 0; kk < 32; kk += 4) {
            v2f a[2], bf[2];
#pragma unroll
            for (int tm = 0; tm < 2; ++tm) {
                int m = wm * 32 + tm * 16 + l15;
                a[tm] = *(const v2f*)&As[m * SA + kk + 2 * half];
            }
#pragma unroll
            for (int tn = 0; tn < 2; ++tn) {
                int nn = wn * 32 + tn * 16 + l15;
                bf[tn] = *(const v2f*)&Bs[nn * SB + kk + 2 * half];
            }
#pragma unroll
            for (int tm = 0; tm < 2; ++tm)
#pragma unroll
                for (int tn = 0; tn < 2; ++tn)
                    acc[tm][tn] = __builtin_amdgcn_wmma_f32_16x16x4_f32(
                        false, a[tm], false, bf[tn], (short)0, acc[tm][tn], false, false);
        }
        __syncthreads();
    }
#pragma unroll
    for (int tm = 0; tm < 2; ++tm)
#pragma unroll
        for (int tn = 0; tn < 2; ++tn) {
            int col = n0 + wn * 32 + tn * 16 + l15;
#pragma unroll
            for (int r = 0; r < 8; ++r) {
                int m = m0 + wm * 32 + tm * 16 + half * 8 + r;
                float v = acc[tm][tn][r];
                if (rowscale) v *= rowscale[m];
                int orow = rowidx ? rowidx[m] : m;
                float* dst = &out[(size_t)orow * HDIM + col];
                if (use_atomic) atomicAdd(dst, v);
                else *dst = v;
            }
        }
}

// ---------------------------------------------------------------------------
extern "C" void kernel_launch(void* const* d_in, const int* in_sizes, int n_in,
                              void* d_out, int out_size, void* d_ws, size_t ws_size,
                              hipStream_t stream) {
    (void)in_sizes; (void)n_in; (void)out_size; (void)ws_size;
    const float* x  = (const float*)d_in[0];
    const float* Wr = (const float*)d_in[1];
    const float* w1 = (const float*)d_in[2];
    const float* w3 = (const float*)d_in[3];
    const float* w2 = (const float*)d_in[4];
    const float* s1 = (const float*)d_in[5];
    const float* s3 = (const float*)d_in[6];
    const float* s2 = (const float*)d_in[7];
    float* out = (float*)d_out;

    char* ws = (char*)d_ws;
    float* score = (float*)ws;                              // E*NTOK f32 (128 KB)
    int*   order = (int*)(ws + (size_t)NEXP * NTOK * 4);    // E*CAP i32 (40 KB)
    float* wsel  = (float*)(ws + (size_t)NEXP * NTOK * 4 + (size_t)NEXP * CAP * 4);
    float* hbuf  = (float*)(ws + (size_t)NEXP * NTOK * 4 + (size_t)NEXP * CAP * 8);
    // hbuf: CAP * IDIM f32 = 15.7 MB  (total ws use ~16 MB)

    hipMemsetAsync(order, 0, (size_t)NEXP * CAP * 4, stream);
    hipMemsetAsync(wsel,  0, (size_t)NEXP * CAP * 4, stream);

    router_kernel<<<NTOK / 256, 256, 0, stream>>>(x, Wr, score);
    select_kernel<<<dim3(NTOK / 256, NEXP), 256, 0, stream>>>(score, order, wsel);

    // shared expert over all tokens, chunked to reuse hbuf; covers every
    // output element with plain stores (runs before routed atomics)
    int done = 0;
    while (done < NTOK) {
        int chunk = NTOK - done; if (chunk > CAP) chunk = CAP;   // 1280,1280,1280,256
        gateup_kernel<<<dim3(IDIM / 128, chunk / 64), 256, 0, stream>>>(
            x + (size_t)done * HDIM, nullptr, s1, s3, hbuf);
        down_kernel<<<dim3(HDIM / 128, chunk / 64), 256, 0, stream>>>(
            hbuf, s2, nullptr, nullptr, out + (size_t)done * HDIM, /*atomic=*/0);
        done += chunk;
    }

    // routed experts: gather via order, scatter-add weighted outputs
    for (int e = 0; e < NEXP; ++e) {
        const float* W1e = w1 + (size_t)e * HDIM * IDIM;
        const float* W3e = w3 + (size_t)e * HDIM * IDIM;
        const float* W2e = w2 + (size_t)e * IDIM * HDIM;
        gateup_kernel<<<dim3(IDIM / 128, CAP / 64), 256, 0, stream>>>(
            x, order + e * CAP, W1e, W3e, hbuf);
        down_kernel<<<dim3(HDIM / 128, CAP / 64), 256, 0, stream>>>(
            hbuf, W2e, order + e * CAP, wsel + e * CAP, out, /*atomic=*/1);
    }
}